// PPLM_Contact_84524956385362
// MI455X (gfx1250) — compile-verified
//
#include <hip/hip_runtime.h>
#include <hip/hip_bf16.h>
#include <math.h>

// ---------------------------------------------------------------------------
// CDNA5 / gfx1250 implementation of the pairformer contact model.
// All matmul-shaped work goes through V_WMMA_F32_16X16X4_F32 (exact fp32 MMA).
// wave32 layouts per CDNA5 ISA:
//   A (16x4 f32): lanes 0-15 row m, VGPR0=K0 VGPR1=K1; lanes 16-31 same rows, K2/K3
//   B (4x16 f32): lanes 0-15 col n, VGPR0=K0 VGPR1=K1; lanes 16-31 same cols, K2/K3
//   C/D (16x16 f32): 8 VGPRs; lane<16 -> M=r, lane>=16 -> M=r+8; N=lane&15
// ---------------------------------------------------------------------------

typedef float v2f __attribute__((ext_vector_type(2)));
typedef float v8f __attribute__((ext_vector_type(8)));

#define LSEQ 128
#define NPIX (LSEQ * LSEQ)   // 16384
#define CCH  64

__device__ __forceinline__ v8f wmma4(v2f a, v2f b, v8f c) {
  // (neg_a, A, neg_b, B, c_mod, C, reuse_a, reuse_b)
  return __builtin_amdgcn_wmma_f32_16x16x4_f32(false, a, false, b, (short)0, c,
                                               false, false);
}

__device__ __forceinline__ float act_apply(float v, int act) {
  switch (act) {
    case 1: return v > 0.f ? v : (expf(v) - 1.f);       // ELU
    case 2: return v > 0.f ? v : 0.f;                   // ReLU
    case 3: return 1.f / (1.f + expf(-v));              // sigmoid
    default: return v;
  }
}

// ---------------------------------------------------------------------------
// Generic WMMA GEMM:  OUT[M,N] = act(A[M,K] * W[K,N] + bias)
// A,W,OUT row-major. M multiple of 16; K even; N arbitrary (masked).
// One wave per 16x16 tile; 4 waves per block. Inner loop is branch-free:
// A fragment is one 8B load, out-of-range columns use a clamped address and a
// 0/1 mask multiply instead of exec-mask predication.
// ---------------------------------------------------------------------------
__global__ void k_gemm(const float* __restrict__ A, const float* __restrict__ W,
                       const float* __restrict__ bias, float* __restrict__ OUT,
                       int M, int N, int K, int act) {
  int tilesN = (N + 15) >> 4;
  int totalTiles = (M >> 4) * tilesN;
  int wave = threadIdx.x >> 5;
  int tile = blockIdx.x * 4 + wave;
  if (tile >= totalTiles) return;
  int tm = tile / tilesN, tn = tile % tilesN;
  int m0 = tm << 4, n0 = tn << 4;
  int lane = threadIdx.x & 31, half = lane >> 4, l16 = lane & 15;
  int row = m0 + l16;
  int col = n0 + l16;
  bool cok = col < N;
  int colC = cok ? col : (N - 1);
  float cmask = cok ? 1.f : 0.f;
  const float* pA = A + (size_t)row * K + half * 2;   // 8B aligned (K even)
  const float* pB = W + colC + (size_t)(half * 2) * N;
  const size_t stepB = (size_t)4 * N;
  int K4 = K & ~3;
  v8f acc = {};
  for (int k = 0; k < K4; k += 4) {
    v2f a = *(const v2f*)pA;
    v2f b;
    b.x = pB[0] * cmask;
    b.y = pB[N] * cmask;
    acc = wmma4(a, b, acc);
    pA += 4;
    pB += stepB;
  }
  if (K4 < K) {  // K remainder (2 for K=338/806); guarded once
    int kr = K4 + half * 2;
    v2f a, b;
    a.x = (kr < K) ? pA[0] : 0.f;
    a.y = (kr + 1 < K) ? pA[1] : 0.f;
    b.x = (kr < K) ? pB[0] * cmask : 0.f;
    b.y = (kr + 1 < K) ? pB[N] * cmask : 0.f;
    acc = wmma4(a, b, acc);
  }
  if (cok) {
    float bb = bias ? bias[col] : 0.f;
    for (int r = 0; r < 8; ++r) {
      int rr = m0 + r + half * 8;
      OUT[(size_t)rr * N + col] = act_apply(acc[r] + bb, act);
    }
  }
}

// ---------------------------------------------------------------------------
// Triangle-mult einsum (per-channel 128x128x128 fp32 GEMMs, channel-last):
//  transpose=0: X[i,j,c] = sum_k P[i,k,c] * Q[k,j,c]
//  transpose=1: X[i,j,c] = sum_k P[i,k,c] * Q[j,k,c]
// One wave per (16x16 (i,j) tile, channel c): 8*8*64 = 4096 tiles.
// Pointer-increment addressing (no per-iter index math).
// ---------------------------------------------------------------------------
__global__ void k_trimul(const float* __restrict__ P, const float* __restrict__ Q,
                         float* __restrict__ X, int transpose) {
  int wave = threadIdx.x >> 5;
  int tile = blockIdx.x * 4 + wave;   // 0..4095
  int c = tile & 63;
  int t2 = tile >> 6;
  int tj = t2 & 7, ti = t2 >> 3;
  int lane = threadIdx.x & 31, half = lane >> 4, l16 = lane & 15;
  int i0 = ti * 16, j0 = tj * 16;
  const float* pA = P + ((size_t)((i0 + l16) * LSEQ + half * 2)) * CCH + c;
  const float* pB;
  size_t stepB, offB;
  if (!transpose) {
    pB = Q + ((size_t)(half * 2 * LSEQ + j0 + l16)) * CCH + c;
    stepB = (size_t)4 * LSEQ * CCH;
    offB = (size_t)LSEQ * CCH;
  } else {
    pB = Q + ((size_t)((j0 + l16) * LSEQ + half * 2)) * CCH + c;
    stepB = 4 * CCH;
    offB = CCH;
  }
  v8f acc = {};
  for (int k = 0; k < LSEQ; k += 4) {
    v2f a, b;
    a.x = pA[0];
    a.y = pA[CCH];
    b.x = pB[0];
    b.y = pB[offB];
    acc = wmma4(a, b, acc);
    pA += 4 * CCH;
    pB += stepB;
  }
  for (int r = 0; r < 8; ++r) {
    int i = i0 + r + half * 8;
    X[((size_t)(i * LSEQ + (j0 + l16))) * CCH + c] = acc[r];
  }
}

// ---------------------------------------------------------------------------
// Fused attention for one (r, h): S = QK^T*scale(+bias); softmax; O = S*V.
// q/k/v/o are [L,L,H*DH] fp32; full 128x128 score matrix in LDS (bank-swizzled).
// modes: 0 cross-row, 1 cross-col, 2 self-row, 3 self-col.
// ---------------------------------------------------------------------------
#define SIDX(r_, c_) ((r_) * 128 + (((c_) + (r_)) & 127))

__global__ void k_attn(const float* __restrict__ q, const float* __restrict__ km,
                       const float* __restrict__ vm, const float* __restrict__ bias,
                       float* __restrict__ o, int mode) {
  __shared__ float S[128 * 128];  // 64 KB
  const int HH = 4, LC = LSEQ * CCH;
  int r = blockIdx.x / HH, h = blockIdx.x % HH;
  size_t qBase, kBase, bBase = 0;
  int qStr, kStr;
  bool useB = (mode < 2);
  if (mode == 0 || mode == 2) { qBase = (size_t)r * LC + h * 16; qStr = CCH; }
  else                        { qBase = (size_t)r * CCH + h * 16; qStr = LC; }
  if (mode == 3)              { kBase = (size_t)r * CCH + h * 16; kStr = LC; }
  else                        { kBase = (size_t)r * LC + h * 16; kStr = CCH; }
  if (useB) bBase = (size_t)r * LSEQ * HH + h;
  int tid = threadIdx.x, wave = tid >> 5, lane = tid & 31;
  int half = lane >> 4, l16 = lane & 15;
  const float scale = 0.25f;  // 1/sqrt(16)

  // Phase 1: S = Q * K^T * scale + bias   (64 tiles, 16 per wave, Kdim=16)
  for (int t = wave; t < 64; t += 4) {
    int m0 = (t >> 3) * 16, n0 = (t & 7) * 16;
    const float* pQ = q + qBase + (size_t)(m0 + l16) * qStr + half * 2;
    const float* pK = km + kBase + (size_t)(n0 + l16) * kStr + half * 2;
    v8f acc = {};
    for (int d = 0; d < 16; d += 4) {
      v2f a = *(const v2f*)(pQ + d);   // contiguous along DH
      v2f b = *(const v2f*)(pK + d);
      acc = wmma4(a, b, acc);
    }
    float bb = useB ? bias[bBase + (size_t)(n0 + l16) * HH] : 0.f;
    for (int rr = 0; rr < 8; ++rr)
      S[SIDX(m0 + rr + half * 8, n0 + l16)] = acc[rr] * scale + bb;
  }
  __syncthreads();

  // Phase 2: row softmax (one row per thread)
  {
    float mx = -3.4e38f;
    for (int k2 = 0; k2 < 128; ++k2) mx = fmaxf(mx, S[SIDX(tid, k2)]);
    float sum = 0.f;
    for (int k2 = 0; k2 < 128; ++k2) {
      float e = expf(S[SIDX(tid, k2)] - mx);
      S[SIDX(tid, k2)] = e;
      sum += e;
    }
    float inv = 1.f / sum;
    for (int k2 = 0; k2 < 128; ++k2) S[SIDX(tid, k2)] *= inv;
  }
  __syncthreads();

  // Phase 3: O = S * V   (M=128, N=16, K=128), 8 tiles / 2 per wave
  for (int t = wave; t < 8; t += 4) {
    int m0 = t * 16;
    const float* pV = vm + kBase + (size_t)(half * 2) * kStr + l16;
    const size_t stepV = (size_t)4 * kStr;
    v8f acc = {};
    for (int k2 = 0; k2 < 128; k2 += 4) {
      int kk = k2 + half * 2;
      v2f a, b;
      a.x = S[SIDX(m0 + l16, kk)];
      a.y = S[SIDX(m0 + l16, kk + 1)];
      b.x = pV[0];
      b.y = pV[kStr];
      acc = wmma4(a, b, acc);
      pV += stepV;
    }
    for (int rr = 0; rr < 8; ++rr)
      o[qBase + (size_t)(m0 + rr + half * 8) * qStr + l16] = acc[rr];
  }
}

// ---------------------------------------------------------------------------
// LayerNorm over last dim (=64) for [rows,64]; one wave32 per row.
// ---------------------------------------------------------------------------
__global__ void k_ln(const float* __restrict__ x, const float* __restrict__ sc,
                     const float* __restrict__ bi, float* __restrict__ y, int rows) {
  int wave = threadIdx.x >> 5;
  int row = blockIdx.x * 4 + wave;
  if (row >= rows) return;
  int lane = threadIdx.x & 31;
  const float* xr = x + (size_t)row * CCH;
  float v0 = xr[lane], v1 = xr[lane + 32];
  float s = v0 + v1;
  for (int off = 16; off > 0; off >>= 1) s += __shfl_xor(s, off, 32);
  float m = s * (1.f / 64.f);
  float d0 = v0 - m, d1 = v1 - m;
  float vv = d0 * d0 + d1 * d1;
  for (int off = 16; off > 0; off >>= 1) vv += __shfl_xor(vv, off, 32);
  float inv = rsqrtf(vv * (1.f / 64.f) + 1e-5f);
  y[(size_t)row * CCH + lane]      = d0 * inv * sc[lane] + bi[lane];
  y[(size_t)row * CCH + lane + 32] = d1 * inv * sc[lane + 32] + bi[lane + 32];
}

// -------------------------- elementwise helpers ----------------------------
__global__ void k_mul(float* __restrict__ d, const float* __restrict__ a,
                      const float* __restrict__ b, int n) {
  int i = blockIdx.x * 256 + threadIdx.x;
  if (i < n) d[i] = a[i] * b[i];
}
__global__ void k_add1(float* __restrict__ z, const float* __restrict__ a, int n) {
  int i = blockIdx.x * 256 + threadIdx.x;
  if (i < n) z[i] += a[i];
}
__global__ void k_add2(float* __restrict__ z, const float* __restrict__ a,
                       const float* __restrict__ b, int n) {
  int i = blockIdx.x * 256 + threadIdx.x;
  if (i < n) z[i] += a[i] + b[i];
}

// build intra pair features: X[pix,0:64]=f[i,:], [64:128]=f[j,:], [128:338]=f2d[:,i,j]
__global__ void k_concat(const float* __restrict__ f, const float* __restrict__ f2d,
                         float* __restrict__ X) {
  size_t idx = (size_t)blockIdx.x * 256 + threadIdx.x;
  if (idx >= (size_t)NPIX * 338) return;
  int pix = (int)(idx / 338), c = (int)(idx % 338);
  int i = pix >> 7, j = pix & 127;
  float v;
  if (c < 64) v = f[i * 64 + c];
  else if (c < 128) v = f[j * 64 + (c - 64)];
  else v = f2d[(size_t)(c - 128) * NPIX + pix];
  X[idx] = v;
}

// channel-first [Cc,L,L] -> channel-last [L*L,Cc]
__global__ void k_transpose_cf(const float* __restrict__ in, float* __restrict__ out,
                               int Cc) {
  size_t idx = (size_t)blockIdx.x * 256 + threadIdx.x;
  if (idx >= (size_t)NPIX * Cc) return;
  int pix = (int)(idx / Cc), t = (int)(idx % Cc);
  out[idx] = in[(size_t)t * NPIX + pix];
}

// [R,Lc] -> [Lc,R]
__global__ void k_transpose2d(const float* __restrict__ in, float* __restrict__ out,
                              int R, int Lc) {
  int idx = blockIdx.x * 256 + threadIdx.x;
  if (idx >= R * Lc) return;
  int l = idx / R, d = idx % R;
  out[idx] = in[d * Lc + l];
}

// 3x3 SAME im2col with optional ELU applied to the source activation.
__global__ void k_im2col(const float* __restrict__ x, float* __restrict__ out,
                         int doElu) {
  size_t idx = (size_t)blockIdx.x * 256 + threadIdx.x;
  if (idx >= (size_t)NPIX * 576) return;
  int pix = (int)(idx / 576), kk = (int)(idx % 576);
  int i = pix >> 7, j = pix & 127;
  int di = kk / 192, r = kk % 192, dj = r / 64, c = r % 64;
  int ii = i + di - 1, jj = j + dj - 1;
  float v = 0.f;
  if (ii >= 0 && ii < LSEQ && jj >= 0 && jj < LSEQ) {
    v = x[((size_t)(ii * LSEQ + jj)) * CCH + c];
    if (doElu) v = v > 0.f ? v : (expf(v) - 1.f);
  }
  out[idx] = v;
}

// ---------------------------------------------------------------------------
// Host orchestration
// ---------------------------------------------------------------------------

// Param leaf offsets (jax tree order: dict keys sorted at every level).
enum { CAP_BIAS_W = 0, CAP_G_B, CAP_G_W, CAP_K_W, CAP_LNM_B, CAP_LNM_S,
       CAP_LNZ_B, CAP_LNZ_S, CAP_OUT_B, CAP_OUT_W, CAP_Q_W, CAP_V_W };      // 12
enum { SAP_G_B = 0, SAP_G_W, SAP_K_W, SAP_LNZ_B, SAP_LNZ_S, SAP_OUT_B,
       SAP_OUT_W, SAP_Q_W, SAP_V_W };                                        // 9
enum { TMP_AG_B = 0, TMP_AG_W, TMP_AP_B, TMP_AP_W, TMP_BG_B, TMP_BG_W,
       TMP_BP_B, TMP_BP_W, TMP_G_B, TMP_G_W, TMP_LNM_B, TMP_LNM_S,
       TMP_LNX_B, TMP_LNX_S, TMP_LNZ_B, TMP_LNZ_S, TMP_OUT_B, TMP_OUT_W };   // 18
enum { TRP_LN_B = 0, TRP_LN_S, TRP_W1_B, TRP_W1_W, TRP_W2_B, TRP_W2_W };     // 6

static const int BLK_CA_L = 0, BLK_CA_R = 12, BLK_SA_L = 24, BLK_SA_R = 33,
                 BLK_TM_L = 42, BLK_TM_R = 60, BLK_TR = 78, BLK_STRIDE = 84;
static const int INTER_BASE = 1008, INTRA_BASE = 1018, LNF_BASE = 1030,
                 OUTP_BASE = 1032;

extern "C" void kernel_launch(void* const* d_in, const int* in_sizes, int n_in,
                              void* d_out, int out_size, void* d_ws, size_t ws_size,
                              hipStream_t stream) {
  (void)in_sizes; (void)n_in; (void)out_size; (void)ws_size;

  const float* in_i1_1d = (const float*)d_in[0];  // [788,128]
  const float* in_i1_2d = (const float*)d_in[1];  // [210,128,128]
  const float* in_i2_1d = (const float*)d_in[2];
  const float* in_i2_2d = (const float*)d_in[3];
  const float* in_x_2d  = (const float*)d_in[4];  // [806,128,128]

  auto P = [&](int i) { return (const float*)d_in[5 + i]; };

  // Workspace layout (floats)
  float* w = (float*)d_ws;
  const size_t TSZ = (size_t)NPIX * CCH;  // 1,048,576 floats (4 MB)
  float* Z  = w;
  float* Z1 = w + TSZ;
  float* Z2 = w + 2 * TSZ;
  float* T[10];
  for (int i = 0; i < 10; ++i) T[i] = w + (3 + i) * TSZ;
  float* BIASB = w + 13 * TSZ;                    // [16384,4]
  float* FBUF  = BIASB + (size_t)NPIX * 4;        // [128,64]
  float* BIG   = FBUF + 128 * 64;                 // up to [16384,806]

  const int EWN = (int)TSZ;
  const int EWG = (EWN + 255) / 256;

  auto gemm = [&](const float* A, const float* Wm, const float* bias, float* O,
                  int M, int N, int K, int act) {
    int tiles = (M >> 4) * ((N + 15) >> 4);
    k_gemm<<<(tiles + 3) / 4, 128, 0, stream>>>(A, Wm, bias, O, M, N, K, act);
  };
  auto lnrm = [&](const float* x, const float* s, const float* b, float* y) {
    k_ln<<<NPIX / 4, 128, 0, stream>>>(x, s, b, y, NPIX);
  };
  auto resnet = [&](float* X, int base) {
    for (int rb = 0; rb < 2; ++rb) {
      int pb = base + rb * 4;  // [c1_b, c1_w, c2_b, c2_w]
      k_im2col<<<(NPIX * 576 + 255) / 256, 256, 0, stream>>>(X, BIG, 1);
      gemm(BIG, P(pb + 1), P(pb + 0), T[0], NPIX, 64, 576, 0);
      k_im2col<<<(NPIX * 576 + 255) / 256, 256, 0, stream>>>(T[0], BIG, 1);
      gemm(BIG, P(pb + 3), P(pb + 2), T[1], NPIX, 64, 576, 0);
      k_add1<<<EWG, 256, 0, stream>>>(X, T[1], EWN);
    }
  };
  auto intra = [&](const float* f1d, const float* f2d, float* Zo) {
    k_transpose2d<<<(788 * 128 + 255) / 256, 256, 0, stream>>>(f1d, BIG, 788, 128);
    gemm(BIG, P(INTRA_BASE + 1), P(INTRA_BASE + 0), FBUF, 128, 64, 788, 0);
    k_concat<<<(NPIX * 338 + 255) / 256, 256, 0, stream>>>(FBUF, f2d, BIG);
    gemm(BIG, P(INTRA_BASE + 3), P(INTRA_BASE + 2), Zo, NPIX, 64, 338, 1);
    resnet(Zo, INTRA_BASE + 4);
  };
  auto tri_mul = [&](const float* m, int pb, int transpose, float* outT) {
    lnrm(Z, P(pb + TMP_LNZ_S), P(pb + TMP_LNZ_B), T[0]);  // zn
    lnrm(m, P(pb + TMP_LNM_S), P(pb + TMP_LNM_B), T[1]);  // mn
    gemm(T[1], P(pb + TMP_AG_W), P(pb + TMP_AG_B), T[2], NPIX, 64, 64, 3);
    gemm(T[1], P(pb + TMP_AP_W), P(pb + TMP_AP_B), T[3], NPIX, 64, 64, 0);
    k_mul<<<EWG, 256, 0, stream>>>(T[2], T[2], T[3], EWN);  // a
    gemm(T[0], P(pb + TMP_BG_W), P(pb + TMP_BG_B), T[4], NPIX, 64, 64, 3);
    gemm(T[0], P(pb + TMP_BP_W), P(pb + TMP_BP_B), T[5], NPIX, 64, 64, 0);
    k_mul<<<EWG, 256, 0, stream>>>(T[4], T[4], T[5], EWN);  // b
    if (!transpose) k_trimul<<<1024, 128, 0, stream>>>(T[2], T[4], T[6], 0);
    else            k_trimul<<<1024, 128, 0, stream>>>(T[4], T[2], T[6], 1);
    lnrm(T[6], P(pb + TMP_LNX_S), P(pb + TMP_LNX_B), T[7]);
    gemm(T[7], P(pb + TMP_OUT_W), P(pb + TMP_OUT_B), T[3], NPIX, 64, 64, 0);
    gemm(T[0], P(pb + TMP_G_W), P(pb + TMP_G_B), T[5], NPIX, 64, 64, 3);
    k_mul<<<EWG, 256, 0, stream>>>(outT, T[3], T[5], EWN);
  };
  auto cross = [&](const float* m, int pb, int mode, float* outT) {
    lnrm(Z, P(pb + CAP_LNZ_S), P(pb + CAP_LNZ_B), T[0]);
    lnrm(m, P(pb + CAP_LNM_S), P(pb + CAP_LNM_B), T[1]);
    gemm(T[0], P(pb + CAP_Q_W), nullptr, T[2], NPIX, 64, 64, 0);
    gemm(T[1], P(pb + CAP_K_W), nullptr, T[3], NPIX, 64, 64, 0);
    gemm(T[1], P(pb + CAP_V_W), nullptr, T[4], NPIX, 64, 64, 0);
    gemm(T[1], P(pb + CAP_BIAS_W), nullptr, BIASB, NPIX, 4, 64, 0);
    k_attn<<<512, 128, 0, stream>>>(T[2], T[3], T[4], BIASB, T[5], mode);
    gemm(T[5], P(pb + CAP_OUT_W), P(pb + CAP_OUT_B), T[6], NPIX, 64, 64, 0);
    gemm(T[0], P(pb + CAP_G_W), P(pb + CAP_G_B), T[7], NPIX, 64, 64, 3);
    k_mul<<<EWG, 256, 0, stream>>>(outT, T[6], T[7], EWN);
  };
  auto selfa = [&](int pb, int mode, float* outT) {
    lnrm(Z, P(pb + SAP_LNZ_S), P(pb + SAP_LNZ_B), T[0]);
    gemm(T[0], P(pb + SAP_Q_W), nullptr, T[2], NPIX, 64, 64, 0);
    gemm(T[0], P(pb + SAP_K_W), nullptr, T[3], NPIX, 64, 64, 0);
    gemm(T[0], P(pb + SAP_V_W), nullptr, T[4], NPIX, 64, 64, 0);
    k_attn<<<512, 128, 0, stream>>>(T[2], T[3], T[4], nullptr, T[5], mode);
    gemm(T[5], P(pb + SAP_OUT_W), P(pb + SAP_OUT_B), T[6], NPIX, 64, 64, 0);
    gemm(T[0], P(pb + SAP_G_W), P(pb + SAP_G_B), T[7], NPIX, 64, 64, 3);
    k_mul<<<EWG, 256, 0, stream>>>(outT, T[6], T[7], EWN);
  };

  // ---- trunk ----
  intra(in_i1_1d, in_i1_2d, Z1);
  intra(in_i2_1d, in_i2_2d, Z2);
  k_transpose_cf<<<(NPIX * 806 + 255) / 256, 256, 0, stream>>>(in_x_2d, BIG, 806);
  gemm(BIG, P(INTER_BASE + 1), P(INTER_BASE + 0), Z, NPIX, 64, 806, 1);
  resnet(Z, INTER_BASE + 2);

  // ---- 12 pairformer blocks ----
  for (int b = 0; b < 12; ++b) {
    int bb = b * BLK_STRIDE;
    tri_mul(Z1, bb + BLK_TM_R, 0, T[8]);
    tri_mul(Z2, bb + BLK_TM_L, 1, T[9]);
    k_add2<<<EWG, 256, 0, stream>>>(Z, T[8], T[9], EWN);
    cross(Z1, bb + BLK_CA_R, 0, T[8]);
    cross(Z2, bb + BLK_CA_L, 1, T[9]);
    k_add2<<<EWG, 256, 0, stream>>>(Z, T[8], T[9], EWN);
    selfa(bb + BLK_SA_R, 2, T[8]);
    selfa(bb + BLK_SA_L, 3, T[9]);
    k_add2<<<EWG, 256, 0, stream>>>(Z, T[8], T[9], EWN);
    // transition
    int tb = bb + BLK_TR;
    lnrm(Z, P(tb + TRP_LN_S), P(tb + TRP_LN_B), T[0]);
    gemm(T[0], P(tb + TRP_W1_W), P(tb + TRP_W1_B), BIG, NPIX, 256, 64, 2);
    gemm(BIG, P(tb + TRP_W2_W), P(tb + TRP_W2_B), T[1], NPIX, 64, 256, 0);
    k_add1<<<EWG, 256, 0, stream>>>(Z, T[1], EWN);
  }

  // ---- head ----
  lnrm(Z, P(LNF_BASE + 1), P(LNF_BASE + 0), T[0]);
  gemm(T[0], P(OUTP_BASE + 1), P(OUTP_BASE + 0), (float*)d_out, NPIX, 1, 64, 3);
}